// Net_18262200943034
// MI455X (gfx1250) — compile-verified
//
#include <hip/hip_runtime.h>

typedef __attribute__((ext_vector_type(2))) float v2f;
typedef __attribute__((ext_vector_type(8))) float v8f;

#define B_SZ   2
#define C_SZ   32
#define H_IN   64
#define W_IN   128
#define D_IN   48      // MAXDISP/4
#define H_OUT  256
#define W_OUT  512
#define D_OUT  192

// ---------------------------------------------------------------------------
// Kernel 1: channel means.  ws[0..16383] = Lm, ws[16384..32767] = Rm
//   Lm[b,h,w] = mean_c left[b,c,h,w]   (layout b*8192 + h*128 + w)
// ---------------------------------------------------------------------------
__global__ __launch_bounds__(256)
void channel_mean_kernel(const float* __restrict__ left,
                         const float* __restrict__ right,
                         float* __restrict__ ws) {
    int t = blockIdx.x * 256 + threadIdx.x;          // 0 .. 32767
    if (t >= 2 * B_SZ * H_IN * W_IN) return;
    int arr = t >> 14;                                // 0 = left, 1 = right
    int rem = t & 16383;
    int b   = rem >> 13;
    int hw  = rem & 8191;
    const float* src = arr ? right : left;
    const float* p = src + b * (C_SZ * H_IN * W_IN) + hw;
    float s = 0.f;
#pragma unroll
    for (int c = 0; c < C_SZ; ++c) s += p[c * (H_IN * W_IN)];
    ws[t] = s * (1.f / (float)C_SZ);
}

// ---------------------------------------------------------------------------
// Kernel 2: per 16-pixel tile: hw-bilinear of implicit cost volume,
// disparity-axis upsample as 192x48x16 f32 WMMA GEMM, online softmax +
// soft-argmin.  One wave (32 lanes) per block.
// ---------------------------------------------------------------------------
__global__ __launch_bounds__(32)
void disparity_regress_kernel(const float* __restrict__ ws,
                              float* __restrict__ out) {
    __shared__ float sL0[W_IN], sL1[W_IN], sR0[W_IN], sR1[W_IN];

    const int lane  = threadIdx.x;        // 0..31
    const int n     = lane & 15;          // pixel within tile
    const int half  = lane >> 4;          // 0: rows 0-7 / K even-pair, 1: rows 8-15
    const int b     = blockIdx.z;
    const int h_out = blockIdx.y;
    const int w_out = blockIdx.x * 16 + n;

    // --- h interpolation (wave-uniform) ---
    const float srch = (float)h_out * (63.f / 255.f);
    const int   h0   = (int)srch;
    const float fh   = srch - (float)h0;
    const int   h1   = min(h0 + 1, H_IN - 1);

    // --- stage the 4 needed mean rows (2 KB) into LDS ---
    const float* Lm = ws + b * (H_IN * W_IN);
    const float* Rm = ws + B_SZ * H_IN * W_IN + b * (H_IN * W_IN);
    for (int i = lane; i < W_IN; i += 32) {
        sL0[i] = Lm[h0 * W_IN + i];
        sL1[i] = Lm[h1 * W_IN + i];
        sR0[i] = Rm[h0 * W_IN + i];
        sR1[i] = Rm[h1 * W_IN + i];
    }
    __syncthreads();

    // --- w interpolation per pixel ---
    const float srcw = (float)w_out * (127.f / 511.f);
    const int   w0   = (int)srcw;
    const float fw   = srcw - (float)w0;
    const int   w1   = min(w0 + 1, W_IN - 1);
    const float wt00 = (1.f - fh) * (1.f - fw) * 0.5f;   // (h0,w0)
    const float wt10 = fh * (1.f - fw) * 0.5f;           // (h1,w0)
    const float wt01 = (1.f - fh) * fw * 0.5f;           // (h0,w1)
    const float wt11 = fh * fw * 0.5f;                   // (h1,w1)

    // --- v[d] in WMMA B-matrix layout: slot [kb][j] holds d = 4*kb + 2*half + j
    float vB[12][2];
#pragma unroll
    for (int kb = 0; kb < 12; ++kb) {
#pragma unroll
        for (int j = 0; j < 2; ++j) {
            const int d = 4 * kb + 2 * half + j;
            float acc = 0.f;
            if (w0 >= d)
                acc += wt00 * (sL0[w0] + sR0[w0 - d]) + wt10 * (sL1[w0] + sR1[w0 - d]);
            if (w1 >= d)
                acc += wt01 * (sL0[w1] + sR0[w1 - d]) + wt11 * (sL1[w1] + sR1[w1 - d]);
            vB[kb][j] = acc;
        }
    }

    // --- X[192] = Wd[192x48] * v[48] via V_WMMA_F32_16X16X4_F32, fused with
    //     online softmax-expectation accumulation ---
    const float sd = 47.f / 191.f;
    float se = 0.f, swe = 0.f;
    for (int mb = 0; mb < 12; ++mb) {
        const int   m_g  = mb * 16 + n;            // this lane's A-matrix row
        const float srcd = (float)m_g * sd;
        const int   i0   = (int)srcd;
        const float fd   = srcd - (float)i0;
        // wave-uniform live K range for this M tile (Wd is 2-diagonal)
        const int k_lo = (int)((float)(mb * 16) * sd);
        const int k_hi = min(D_IN - 1, (int)((float)(mb * 16 + 15) * sd) + 1);
        const int kb_lo = k_lo >> 2, kb_hi = k_hi >> 2;

        v8f acc = {};
#pragma unroll
        for (int kb = 0; kb < 12; ++kb) {
            if (kb >= kb_lo && kb <= kb_hi) {       // uniform guard; EXEC stays all-1
                const int k0 = kb * 4 + 2 * half;   // A layout: VGPR0->K=2h, VGPR1->K=2h+1
                const int k1 = k0 + 1;
                const float a0 = (k0 == i0) ? (1.f - fd) : ((k0 == i0 + 1) ? fd : 0.f);
                const float a1 = (k1 == i0) ? (1.f - fd) : ((k1 == i0 + 1) ? fd : 0.f);
                v2f A;  A.x  = a0;         A.y  = a1;
                v2f Bv; Bv.x = vB[kb][0];  Bv.y = vB[kb][1];
                acc = __builtin_amdgcn_wmma_f32_16x16x4_f32(
                        false, A, false, Bv, (short)0, acc, false, false);
            }
        }
        // D tile: VGPR r, lanes 0-15 -> row r, lanes 16-31 -> row r+8
#pragma unroll
        for (int r = 0; r < 8; ++r) {
            const float x  = acc[r];
            const float dd = (float)(mb * 16 + r + 8 * half);
            const float e  = __expf(x);
            se  += e;
            swe += e * dd;
        }
    }

    // fold the two half-wave partial sums (same pixel lives in lane n and n+16)
    se  += __shfl_xor(se, 16, 32);
    swe += __shfl_xor(swe, 16, 32);
    if (half == 0)
        out[(b * H_OUT + h_out) * W_OUT + w_out] = swe / se;
}

// ---------------------------------------------------------------------------
extern "C" void kernel_launch(void* const* d_in, const int* in_sizes, int n_in,
                              void* d_out, int out_size, void* d_ws, size_t ws_size,
                              hipStream_t stream) {
    const float* left  = (const float*)d_in[0];
    const float* right = (const float*)d_in[1];
    float* out = (float*)d_out;
    float* ws  = (float*)d_ws;          // needs 2*2*64*128 floats = 128 KB

    channel_mean_kernel<<<dim3(128), dim3(256), 0, stream>>>(left, right, ws);
    disparity_regress_kernel<<<dim3(W_OUT / 16, H_OUT, B_SZ), dim3(32), 0, stream>>>(ws, out);
}